// MLP_64450279244218
// MI455X (gfx1250) — compile-verified
//
#include <hip/hip_runtime.h>
#include <hip/hip_bf16.h>

// Problem constants
#define NIN   1600
#define NH1   64
#define NH2   32
#define NW    512
#define LDSTR 68   // padded LDS row stride (floats): 68*4B=272B, 16B aligned, conflict-free

typedef float v2f __attribute__((ext_vector_type(2)));
typedef float v8f __attribute__((ext_vector_type(8)));

// ---------------------------------------------------------------------------
// Kernel 1: A = x @ W1 + b1   (512 x 64),  B = y @ W1   (512 x 64)
// grid = 512 blocks, block = 64 threads. W1 coalesced across threads,
// x/y row element broadcast across the block.
// ---------------------------------------------------------------------------
__global__ void mlp_precompute_kernel(const float* __restrict__ x,
                                      const float* __restrict__ y,
                                      const float* __restrict__ W1,
                                      const float* __restrict__ b1,
                                      float* __restrict__ A,
                                      float* __restrict__ B) {
    const int row = blockIdx.x;      // 0..511
    const int col = threadIdx.x;     // 0..63
    const float* __restrict__ xr = x + row * NIN;
    const float* __restrict__ yr = y + row * NIN;
    float a = 0.0f, b = 0.0f;
    #pragma unroll 8
    for (int i = 0; i < NIN; ++i) {
        const float w = W1[i * NH1 + col];
        a = fmaf(xr[i], w, a);
        b = fmaf(yr[i], w, b);
    }
    A[row * NH1 + col] = a + b1[col];
    B[row * NH1 + col] = b;
}

// ---------------------------------------------------------------------------
// Kernel 2: pairwise MLP using V_WMMA_F32_16X16X4_F32 for layer 2.
// Block = 256 threads (8 waves), tile = 16 w x 16 q outputs.
// Each wave handles 2 w-rows; per w-row: D(16q x 32) = relu(A[w]-B[q]) @ W2
// with b2 pre-loaded into the accumulator, K=64 as 16 chained K=4 WMMAs,
// N=32 as two 16-wide halves. Layer 3 = per-lane FMA + shfl_xor reduction.
// ---------------------------------------------------------------------------
__global__ void mlp_pairwise_wmma_kernel(const float* __restrict__ A,
                                         const float* __restrict__ B,
                                         const float* __restrict__ W2,
                                         const float* __restrict__ b2,
                                         const float* __restrict__ W3,
                                         const float* __restrict__ b3,
                                         float* __restrict__ out) {
    __shared__ float As[16 * LDSTR];
    __shared__ float Bs[16 * LDSTR];

    const int w0   = blockIdx.x * 16;
    const int q0   = blockIdx.y * 16;
    const int tid  = threadIdx.x;       // 0..255
    const int lane = tid & 31;
    const int wave = tid >> 5;          // 0..7
    const int half = lane >> 4;         // 0 or 1 (K sub-pair / M-half selector)
    const int l16  = lane & 15;         // M (q index) for A-frag, N for B-frag

    // Cooperative stage of A (16 w-rows) and B (16 q-rows) into LDS.
    {
        const int r  = tid >> 4;         // 0..15 row
        const int c4 = (tid & 15) * 4;   // 0..60 col (float4 granular)
        const float4 av = *(const float4*)(A + (w0 + r) * NH1 + c4);
        const float4 bv = *(const float4*)(B + (q0 + r) * NH1 + c4);
        *(float4*)&As[r * LDSTR + c4] = av;
        *(float4*)&Bs[r * LDSTR + c4] = bv;
    }
    __syncthreads();

    // W2 fragments held in registers: wb0 = N 0..15, wb1 = N 16..31.
    // B-matrix 4x16 layout: lanes 0-15 hold K={0,1}, lanes 16-31 hold K={2,3},
    // N = lane%16. Chunk c covers global K = 4c .. 4c+3.
    v2f wb0[16], wb1[16];
    #pragma unroll
    for (int c = 0; c < 16; ++c) {
        const int k0 = c * 4 + half * 2;
        wb0[c].x = W2[(k0    ) * NH2 +      l16];
        wb0[c].y = W2[(k0 + 1) * NH2 +      l16];
        wb1[c].x = W2[(k0    ) * NH2 + 16 + l16];
        wb1[c].y = W2[(k0 + 1) * NH2 + 16 + l16];
    }
    const float b2a = b2[l16];
    const float b2b = b2[16 + l16];
    const float w3a = W3[l16];
    const float w3b = W3[16 + l16];
    const float bias3 = b3[0];

    #pragma unroll
    for (int s = 0; s < 2; ++s) {
        const int wl = wave * 2 + s;     // local w row 0..15
        // Accumulators initialized with b2 broadcast (C layout: VGPR r,
        // lane -> (M = r + 8*half, N = l16 [+16 for second half]))
        v8f acc0, acc1;
        #pragma unroll
        for (int r = 0; r < 8; ++r) { acc0[r] = b2a; acc1[r] = b2b; }

        #pragma unroll
        for (int c = 0; c < 16; ++c) {
            const int k0 = c * 4 + half * 2;
            // A-frag (16x4 f32): lane l16 = M = q index; VGPR0/1 = K = k0,k0+1
            const float a0  = As[wl  * LDSTR + k0];
            const float a1  = As[wl  * LDSTR + k0 + 1];
            const float bq0 = Bs[l16 * LDSTR + k0];
            const float bq1 = Bs[l16 * LDSTR + k0 + 1];
            v2f af;
            af.x = fmaxf(a0 - bq0, 0.0f);   // h1 = relu(A[w] - B[q]) fused
            af.y = fmaxf(a1 - bq1, 0.0f);
            acc0 = __builtin_amdgcn_wmma_f32_16x16x4_f32(
                false, af, false, wb0[c], (short)0, acc0, false, false);
            acc1 = __builtin_amdgcn_wmma_f32_16x16x4_f32(
                false, af, false, wb1[c], (short)0, acc1, false, false);
        }

        // Layer 3: o[m] = relu(sum_n relu(D[m,n]) * W3[n] + b3)
        float t[8];
        #pragma unroll
        for (int r = 0; r < 8; ++r) {
            const float h20 = fmaxf(acc0[r], 0.0f);
            const float h21 = fmaxf(acc1[r], 0.0f);
            t[r] = fmaf(h20, w3a, h21 * w3b);
        }
        const int w = w0 + wl;
        #pragma unroll
        for (int r = 0; r < 8; ++r) {
            float v = t[r];
            v += __shfl_xor(v, 1, 32);
            v += __shfl_xor(v, 2, 32);
            v += __shfl_xor(v, 4, 32);
            v += __shfl_xor(v, 8, 32);      // sum within each 16-lane half
            v = fmaxf(v + bias3, 0.0f);
            // Row m = r + 8*half is replicated in all 16 lanes of the half;
            // one lane per half stores it.
            if (l16 == r) {
                out[w * NW + q0 + r + 8 * half] = v;
            }
        }
    }
}

// ---------------------------------------------------------------------------
extern "C" void kernel_launch(void* const* d_in, const int* in_sizes, int n_in,
                              void* d_out, int out_size, void* d_ws, size_t ws_size,
                              hipStream_t stream) {
    const float* x  = (const float*)d_in[0];
    const float* y  = (const float*)d_in[1];
    const float* W1 = (const float*)d_in[2];
    const float* b1 = (const float*)d_in[3];
    const float* W2 = (const float*)d_in[4];
    const float* b2 = (const float*)d_in[5];
    const float* W3 = (const float*)d_in[6];
    const float* b3 = (const float*)d_in[7];
    float* out = (float*)d_out;

    float* A = (float*)d_ws;                 // 512*64 floats = 128 KB
    float* B = A + NW * NH1;                 // 512*64 floats = 128 KB

    mlp_precompute_kernel<<<dim3(NW), dim3(NH1), 0, stream>>>(x, y, W1, b1, A, B);

    dim3 grid(NW / 16, NW / 16);             // 32 x 32 blocks
    mlp_pairwise_wmma_kernel<<<grid, dim3(256), 0, stream>>>(A, B, W2, b2, W3, b3, out);
}